// GraphConvLayer_76493367542297
// MI455X (gfx1250) — compile-verified
//
#include <hip/hip_runtime.h>
#include <hip/hip_bf16.h>

// ---------------------------------------------------------------------------
// GraphConv fused kernel for MI455X (gfx1250, wave32).
//   neigh_mean = mean_k x[adj[b,n,k]]          (gather, L2-resident)
//   h = [x | neigh_mean] @ W + b               (WMMA f32 16x16x4)
//   y = LayerNorm(gelu(h) + x) * gamma + beta
// Sizes fixed by the reference.
// ---------------------------------------------------------------------------

#define GB_B 4
#define GB_N 8192
#define GB_D 128
#define GB_K 32
#define ROWS 64            // rows per workgroup (4 waves * 16)
#define HPITCH 260         // 256 + 4 pad -> conflict-free A-fragment ds_load_b64
#define YPITCH 132         // 128 + 4 pad -> conflict-free epilogue stores
#define LN_EPS 1e-5f

typedef __attribute__((ext_vector_type(2))) float v2f;
typedef __attribute__((ext_vector_type(8))) float v8f;

__global__ __launch_bounds__(128)
void graphconv_fused_kernel(const float* __restrict__ x,
                            const long long* __restrict__ adj,
                            const float* __restrict__ W,
                            const float* __restrict__ bias,
                            const float* __restrict__ gamma,
                            const float* __restrict__ beta,
                            float* __restrict__ out)
{
    __shared__ float sh_h[ROWS * HPITCH];   // [x(0..127) | mean(128..255)] per row
    __shared__ float sh_y[ROWS * YPITCH];   // post-GELU + residual
    __shared__ int   sh_adj[ROWS * GB_K];

    const int tid   = threadIdx.x;
    const int gbase = blockIdx.x * ROWS;          // flattened row base over B*N
    const int batch = gbase / GB_N;               // N % ROWS == 0 -> uniform
    const int nbase = gbase % GB_N;

    const float*      xb   = x   + (size_t)batch * GB_N * GB_D;
    const long long*  adjb = adj + ((size_t)batch * GB_N + nbase) * GB_K;

    // ---------------- Phase A: stage adj (int32) + x rows into LDS ----------
    for (int i = tid; i < ROWS * GB_K; i += 128)
        sh_adj[i] = (int)adjb[i];

    for (int i = tid; i < ROWS * (GB_D / 4); i += 128) {
        const int r  = i >> 5;            // 32 float4 groups per row
        const int fg = i & 31;
        const float4 v = *(const float4*)(xb + (size_t)(nbase + r) * GB_D + 4 * fg);
        float* dst = &sh_h[r * HPITCH + 4 * fg];
        dst[0] = v.x; dst[1] = v.y; dst[2] = v.z; dst[3] = v.w;
    }
    __syncthreads();

    // ---------------- Phase B: gather + mean over K=32 neighbors ------------
    for (int i = tid; i < ROWS * (GB_D / 4); i += 128) {
        const int r  = i >> 5;
        const int fg = i & 31;
        const int* arow = &sh_adj[r * GB_K];
        float ax = 0.f, ay = 0.f, az = 0.f, aw = 0.f;
        #pragma unroll 8
        for (int k = 0; k < GB_K; ++k) {
            const int idx = arow[k];
            const float4 v = *(const float4*)(xb + (size_t)idx * GB_D + 4 * fg);
            ax += v.x; ay += v.y; az += v.z; aw += v.w;
        }
        const float inv = 1.0f / (float)GB_K;
        float* dst = &sh_h[r * HPITCH + GB_D + 4 * fg];
        dst[0] = ax * inv; dst[1] = ay * inv; dst[2] = az * inv; dst[3] = aw * inv;
    }
    __syncthreads();

    // ---------------- Phase C: WMMA GEMM  h(16x256) @ W(256x128) ------------
    const int wv   = tid >> 5;          // wave id (0..3) -> rows 16*wv..16*wv+15
    const int lane = tid & 31;
    const int half = lane >> 4;         // A lanes 0-15: K=0,1 ; 16-31: K=2,3
    const int m    = lane & 15;         // A: M index / B,C,D: N index
    const int rowbase = wv * 16;

    v8f acc[8];
    #pragma unroll
    for (int nt = 0; nt < 8; ++nt) acc[nt] = (v8f){0.f,0.f,0.f,0.f,0.f,0.f,0.f,0.f};

    #pragma unroll 4
    for (int t = 0; t < 64; ++t) {                 // K chunks of 4
        const int krow = 4 * t + 2 * half;
        // A fragment: h[rowbase+m][krow], h[rowbase+m][krow+1]  (8B aligned)
        const v2f a = *(const v2f*)&sh_h[(rowbase + m) * HPITCH + krow];
        const float* wrow = W + (size_t)krow * GB_D + m;
        #pragma unroll
        for (int nt = 0; nt < 8; ++nt) {
            v2f bf;
            bf.x = wrow[16 * nt];                  // B[krow  ][16*nt+m]
            bf.y = wrow[GB_D + 16 * nt];           // B[krow+1][16*nt+m]
            acc[nt] = __builtin_amdgcn_wmma_f32_16x16x4_f32(
                false, a, false, bf, (short)0, acc[nt], false, false);
        }
    }

    // ---------------- epilogue: +bias, exact GELU, residual -> sh_y ---------
    #pragma unroll
    for (int nt = 0; nt < 8; ++nt) {
        const int col = 16 * nt + m;
        const float bcol = bias[col];
        #pragma unroll
        for (int v = 0; v < 8; ++v) {
            const int rloc = rowbase + v + 8 * half;    // D element (v+8*half, m)
            float val = acc[nt][v] + bcol;
            float g   = 0.5f * val * (1.0f + erff(val * 0.70710678118654752f));
            sh_y[rloc * YPITCH + col] = g + sh_h[rloc * HPITCH + col]; // + x residual
        }
    }
    __syncthreads();

    // ---------------- Phase D: LayerNorm per row (wave32 reduction) ---------
    const float g0 = gamma[4 * lane + 0], g1 = gamma[4 * lane + 1];
    const float g2 = gamma[4 * lane + 2], g3 = gamma[4 * lane + 3];
    const float b0 = beta[4 * lane + 0],  b1 = beta[4 * lane + 1];
    const float b2 = beta[4 * lane + 2],  b3 = beta[4 * lane + 3];

    for (int r = rowbase; r < rowbase + 16; ++r) {
        const float* yr = &sh_y[r * YPITCH + 4 * lane];
        const float v0 = yr[0], v1 = yr[1], v2 = yr[2], v3 = yr[3];
        float s  = v0 + v1 + v2 + v3;
        float ss = v0 * v0 + v1 * v1 + v2 * v2 + v3 * v3;
        #pragma unroll
        for (int off = 16; off > 0; off >>= 1) {
            s  += __shfl_xor(s,  off, 32);
            ss += __shfl_xor(ss, off, 32);
        }
        const float mu  = s * (1.0f / GB_D);
        const float var = ss * (1.0f / GB_D) - mu * mu;
        const float rs  = rsqrtf(var + LN_EPS);

        float4 o;
        o.x = (v0 - mu) * rs * g0 + b0;
        o.y = (v1 - mu) * rs * g1 + b1;
        o.z = (v2 - mu) * rs * g2 + b2;
        o.w = (v3 - mu) * rs * g3 + b3;
        *(float4*)(out + ((size_t)gbase + r) * GB_D + 4 * lane) = o;
    }
}

extern "C" void kernel_launch(void* const* d_in, const int* in_sizes, int n_in,
                              void* d_out, int out_size, void* d_ws, size_t ws_size,
                              hipStream_t stream) {
    (void)in_sizes; (void)n_in; (void)out_size; (void)d_ws; (void)ws_size;
    const float*     x     = (const float*)d_in[0];
    const long long* adj   = (const long long*)d_in[1];   // int64
    const float*     W     = (const float*)d_in[2];
    const float*     bias  = (const float*)d_in[3];
    const float*     gamma = (const float*)d_in[4];
    const float*     beta  = (const float*)d_in[5];
    float*           out   = (float*)d_out;

    const int blocks = (GB_B * GB_N) / ROWS;   // 512
    graphconv_fused_kernel<<<blocks, 128, 0, stream>>>(x, adj, W, bias, gamma, beta, out);
}